// DetectionLoss_11304353923123
// MI455X (gfx1250) — compile-verified
//
#include <hip/hip_runtime.h>

// ---------------- problem constants (match reference) ----------------
#define NCLS   3
#define TT     20          // targets per image
#define NBATCH 64
#define MAXA   9408        // anchors at the largest scale (56*56*3)
#define NTHR   256         // 8 waves of 32

typedef __attribute__((ext_vector_type(2))) float v2f;
typedef __attribute__((ext_vector_type(8))) float v8f;

// ---------------------------------------------------------------------
// Wave-level sum using V_WMMA_F32_16X16X4_F32.
// A(16x4): lane L<16 holds (M=L,K=0) in a[0], (M=L,K=1) in a[1];
//          lane L>=16 holds (M=L-16,K=2),(M=L-16,K=3).
// With B = all ones, D(m,n) = sum_k A(m,k) for every n, so lane L's
// 8 accumulators hold rows {L-half}; summing them and adding the
// opposite half (shfl_xor 16) yields the full 32-lane sum in all lanes.
// EXEC must be all ones: only call uniformly from a full block.
// ---------------------------------------------------------------------
__device__ __forceinline__ float wave_sum_wmma(float v) {
  v2f a;  a[0] = v;    a[1] = 0.0f;
  v2f bm; bm[0] = 1.0f; bm[1] = 1.0f;
  v8f c = {0.f, 0.f, 0.f, 0.f, 0.f, 0.f, 0.f, 0.f};
  c = __builtin_amdgcn_wmma_f32_16x16x4_f32(false, a, false, bm,
                                            (short)0, c, false, false);
  float s = c[0] + c[1] + c[2] + c[3] + c[4] + c[5] + c[6] + c[7];
  s += __shfl_xor(s, 16, 32);   // combine the two 16-row halves (wave32)
  return s;
}

__device__ __forceinline__ float block_sum(float v, float* red) {
  float ws = wave_sum_wmma(v);
  int lane = threadIdx.x & 31;
  int wid  = threadIdx.x >> 5;
  __syncthreads();                    // protect red[] from previous use
  if (lane == 0) red[wid] = ws;
  __syncthreads();
  float r = 0.f;
#pragma unroll
  for (int k = 0; k < NTHR / 32; ++k) r += red[k];
  return r;                           // every thread gets the block sum
}

__device__ __forceinline__ float bce_logit(float x, float y) {
  return fmaxf(x, 0.f) - x * y + log1pf(expf(-fabsf(x)));
}

// ---------------------------------------------------------------------
// One block per (image b = blockIdx.x, scale s = blockIdx.y).
// ---------------------------------------------------------------------
__global__ __launch_bounds__(NTHR) void det_loss_main(
    const float* __restrict__ pred1, const float* __restrict__ pred2,
    const float* __restrict__ pred3, const float* __restrict__ anc1,
    const float* __restrict__ anc2,  const float* __restrict__ anc3,
    const float* __restrict__ tboxes, const int* __restrict__ tlabels,
    float* __restrict__ acc) {
  const int b = blockIdx.x;
  const int s = blockIdx.y;
  const float* pred;
  const float* anc;
  int H;
  if (s == 0)      { pred = pred1; anc = anc1; H = 56; }
  else if (s == 1) { pred = pred2; anc = anc2; H = 28; }
  else             { pred = pred3; anc = anc3; H = 14; }
  const int W = H;
  const int A = H * W * 3;

  __shared__ float tb[TT][4];
  __shared__ float tarea[TT];
  __shared__ int   tl[TT];
  __shared__ unsigned long long bestkey[TT];
  __shared__ unsigned char st[MAXA];       // bits[4:0]=matched_t, 32=pos, 64=neg
  __shared__ unsigned negkeys[MAXA];       // compacted BCE bit-keys of negatives
  __shared__ unsigned hist[256];
  __shared__ float red[NTHR / 32];
  __shared__ int negcnt, sh_d, sh_rem;

  const int tid = threadIdx.x;
  if (tid < TT) {
    float x0 = tboxes[((b * TT) + tid) * 4 + 0];
    float y0 = tboxes[((b * TT) + tid) * 4 + 1];
    float x1 = tboxes[((b * TT) + tid) * 4 + 2];
    float y1 = tboxes[((b * TT) + tid) * 4 + 3];
    tb[tid][0] = x0; tb[tid][1] = y0; tb[tid][2] = x1; tb[tid][3] = y1;
    tarea[tid] = (x1 - x0) * (y1 - y0);
    tl[tid] = tlabels[b * TT + tid];
    bestkey[tid] = 0ull;
  }
  if (tid == 0) negcnt = 0;
  __syncthreads();

  // ---------------- Phase A: IoU, per-anchor & per-target argmax ------
  unsigned long long lb[TT];
#pragma unroll
  for (int t = 0; t < TT; ++t) lb[t] = 0ull;

  for (int i0 = 0; i0 < A; i0 += NTHR) {
    const int i = i0 + tid;
    const bool valid = (i < A);
    const int ii = valid ? i : 0;
    const float ax0 = anc[ii * 4 + 0], ay0 = anc[ii * 4 + 1];
    const float ax1 = anc[ii * 4 + 2], ay1 = anc[ii * 4 + 3];
    const float aarea = (ax1 - ax0) * (ay1 - ay0);
    float best = -1.f;
    int bt = 0;
#pragma unroll
    for (int t = 0; t < TT; ++t) {
      float lx = fmaxf(ax0, tb[t][0]);
      float ly = fmaxf(ay0, tb[t][1]);
      float rx = fminf(ax1, tb[t][2]);
      float ry = fminf(ay1, tb[t][3]);
      float iw = fmaxf(rx - lx, 0.f);
      float ih = fmaxf(ry - ly, 0.f);
      float inter = iw * ih;
      float iou = inter / (aarea + tarea[t] - inter + 1e-9f);
      if (iou > best) { best = iou; bt = t; }   // first-max like jnp.argmax
      if (valid) {
        // larger iou wins; ties -> smaller anchor index (first occurrence)
        unsigned long long key =
            ((unsigned long long)__float_as_uint(iou) << 32) |
            (unsigned)(~(unsigned)i);
        if (key > lb[t]) lb[t] = key;
      }
    }
    if (valid) {
      unsigned char fl = (unsigned char)bt;
      if (best >= 0.5f) fl |= 32;
      if (best < 0.3f)  fl |= 64;
      st[i] = fl;
    }
  }
#pragma unroll
  for (int t = 0; t < TT; ++t) atomicMax(&bestkey[t], lb[t]);
  __syncthreads();

  // ---------------- Phase B: best-anchor scatter overrides ------------
  if (tid == 0) {
    for (int t = 0; t < TT; ++t) {              // sequential: last t wins
      unsigned ai = ~(unsigned)(bestkey[t] & 0xFFFFFFFFull);
      st[ai] = (unsigned char)t | 32u;          // pos=1, neg=0, matched=t
    }
  }
  __syncthreads();

  // ---------------- Phase C: per-anchor losses ------------------------
  const size_t chs = (size_t)H * W;
  const float* pb = pred + (size_t)b * 24 * chs;
  float n_pos = 0.f, s_pbce = 0.f, s_ce = 0.f, s_sl1 = 0.f;

  for (int i0 = 0; i0 < A; i0 += NTHR) {
    const int i = i0 + tid;
    const bool valid = (i < A);
    const int ii = valid ? i : 0;
    const unsigned fl = valid ? (unsigned)st[ii] : 0u;
    const bool pos = (fl & 32u) != 0u;
    const bool neg = (fl & 64u) != 0u;
    const int mt = (int)(fl & 31u);
    const int cell = ii / 3, a = ii % 3;
    const int hh = cell / W, ww = cell % W;
    const float* pc = pb + (size_t)(a * 8) * chs + (size_t)hh * W + ww;
    const float x = valid ? pc[4 * chs] : 0.f;
    const float bce = bce_logit(x, pos ? 1.f : 0.f);
    if (pos) {
      n_pos  += 1.f;
      s_pbce += bce;
      // class cross-entropy (3 classes)
      float c0 = pc[5 * chs], c1 = pc[6 * chs], c2 = pc[7 * chs];
      float m = fmaxf(c0, fmaxf(c1, c2));
      float lse = m + logf(expf(c0 - m) + expf(c1 - m) + expf(c2 - m));
      int tgt = tl[mt] - 1;
      float ct = (tgt == 0) ? c0 : ((tgt == 1) ? c1 : c2);
      s_ce += lse - ct;
      // smooth-L1 on encoded deltas
      float ax0 = anc[ii * 4 + 0], ay0 = anc[ii * 4 + 1];
      float ax1 = anc[ii * 4 + 2], ay1 = anc[ii * 4 + 3];
      float aw = ax1 - ax0, ah = ay1 - ay0;
      float acx = (ax0 + ax1) * 0.5f, acy = (ay0 + ay1) * 0.5f;
      float g0 = tb[mt][0], g1 = tb[mt][1], g2 = tb[mt][2], g3 = tb[mt][3];
      float gw = g2 - g0, gh = g3 - g1;
      float gcx = (g0 + g2) * 0.5f, gcy = (g1 + g3) * 0.5f;
      float td0 = (gcx - acx) / aw, td1 = (gcy - acy) / ah;
      float td2 = logf(gw / aw),   td3 = logf(gh / ah);
      float pd0 = pc[0], pd1 = pc[chs], pd2 = pc[2 * chs], pd3 = pc[3 * chs];
      float d;
      d = fabsf(pd0 - td0); s_sl1 += (d < 1.f) ? 0.5f * d * d : d - 0.5f;
      d = fabsf(pd1 - td1); s_sl1 += (d < 1.f) ? 0.5f * d * d : d - 0.5f;
      d = fabsf(pd2 - td2); s_sl1 += (d < 1.f) ? 0.5f * d * d : d - 0.5f;
      d = fabsf(pd3 - td3); s_sl1 += (d < 1.f) ? 0.5f * d * d : d - 0.5f;
    }
    // wave-aggregated compaction of negative BCE keys into LDS
    unsigned key  = __float_as_uint(bce);   // bce >= 0 -> monotone bits
    unsigned mask = (unsigned)__ballot(neg);
    int lane = tid & 31;
    int base = 0;
    if (lane == 0) base = atomicAdd(&negcnt, __popc(mask));
    base = __shfl(base, 0, 32);
    if (neg) negkeys[base + __popc(mask & ((1u << lane) - 1u))] = key;
  }
  __syncthreads();
  const int num_neg = negcnt;

  float NP = block_sum(n_pos,  red);
  float SB = block_sum(s_pbce, red);
  float SC = block_sum(s_ce,   red);
  float SL = block_sum(s_sl1,  red);
  const int num_pos = (int)(NP + 0.5f);
  const int need = min(3 * num_pos, num_neg);

  // ---------------- Phase D: exact top-`need` negative BCE sum --------
  float topk = 0.f;
  if (need > 0) {                               // uniform across block
    unsigned prefix = 0u;
    int remaining = need;
    for (int shift = 24; shift >= 0; shift -= 8) {
      if (tid < 256) hist[tid] = 0u;
      __syncthreads();
      const unsigned hm = (shift == 24) ? 0u : (0xFFFFFFFFu << (shift + 8));
      for (int j = tid; j < num_neg; j += NTHR) {
        unsigned key = negkeys[j];
        if (((key ^ prefix) & hm) == 0u)
          atomicAdd(&hist[(key >> shift) & 255u], 1u);
      }
      __syncthreads();
      if (tid == 0) {
        int c = 0, d = 255;
        for (; d > 0; --d) {
          int hc = (int)hist[d];
          if (c + hc >= remaining) break;
          c += hc;
        }
        sh_d = d;
        sh_rem = remaining - c;
      }
      __syncthreads();
      prefix |= ((unsigned)sh_d) << shift;
      remaining = sh_rem;
      __syncthreads();
    }
    // prefix == need-th largest key; sum strictly-greater + tie fill
    float sg = 0.f, cg = 0.f;
    for (int j = tid; j < num_neg; j += NTHR) {
      unsigned key = negkeys[j];
      if (key > prefix) { sg += __uint_as_float(key); cg += 1.f; }
    }
    float SG = block_sum(sg, red);
    float CG = block_sum(cg, red);
    topk = SG + (float)(need - (int)(CG + 0.5f)) * __uint_as_float(prefix);
  }

  if (tid == 0) {
    float obj_l = (SB + topk) / (float)max(num_pos + need, 1);
    float cls_l = SC / (float)max(num_pos, 1);
    float loc_l = SL / (float)max(4 * num_pos, 1);
    atomicAdd(&acc[0], obj_l);
    atomicAdd(&acc[1], cls_l);
    atomicAdd(&acc[2], loc_l);
  }
}

__global__ void det_loss_init(float* acc) {
  if (threadIdx.x < 3) acc[threadIdx.x] = 0.f;
}

__global__ void det_loss_final(const float* __restrict__ acc,
                               float* __restrict__ out) {
  if (threadIdx.x == 0) {
    float o = acc[0] / (float)NBATCH;
    float c = acc[1] / (float)NBATCH;
    float l = acc[2] / (float)NBATCH;
    out[0] = o;
    out[1] = c;
    out[2] = l;
    out[3] = o + c + 2.f * l;
  }
}

extern "C" void kernel_launch(void* const* d_in, const int* in_sizes, int n_in,
                              void* d_out, int out_size, void* d_ws,
                              size_t ws_size, hipStream_t stream) {
  (void)in_sizes; (void)n_in; (void)out_size; (void)ws_size;
  const float* pred1  = (const float*)d_in[0];
  const float* pred2  = (const float*)d_in[1];
  const float* pred3  = (const float*)d_in[2];
  const float* anc1   = (const float*)d_in[3];
  const float* anc2   = (const float*)d_in[4];
  const float* anc3   = (const float*)d_in[5];
  const float* tboxes = (const float*)d_in[6];
  const int*   tlabls = (const int*)d_in[7];
  float* acc = (float*)d_ws;

  det_loss_init<<<1, 32, 0, stream>>>(acc);
  dim3 grid(NBATCH, 3);
  det_loss_main<<<grid, NTHR, 0, stream>>>(pred1, pred2, pred3, anc1, anc2,
                                           anc3, tboxes, tlabls, acc);
  det_loss_final<<<1, 32, 0, stream>>>(acc, (float*)d_out);
}